// GCN_47107201303133
// MI455X (gfx1250) — compile-verified
//
#include <hip/hip_runtime.h>

#define N_NODES 50000
#define N_EDGES 1600000
#define D_HID   128
#define D_OUT   64
#define BN_EPS  1e-5f

typedef __attribute__((ext_vector_type(2))) float v2f;
typedef __attribute__((ext_vector_type(4))) float v4f;
typedef __attribute__((ext_vector_type(8))) float v8f;

// ---------------------------------------------------------------------------
// C[16*gridDim.x x DN] = A[... x 128] @ W[128 x DN], fp32 via V_WMMA_F32_16X16X4_F32
// block = 32 * (DN/16) threads; wave w computes output columns [w*16, w*16+16)
// for the 16-row tile mbase = blockIdx.x*16.  EXEC all-ones (no divergence).
// ---------------------------------------------------------------------------
template <int DN>
__global__ __launch_bounds__(32 * (DN / 16)) void gemm_f32_wmma(
    const float* __restrict__ A, const float* __restrict__ W,
    float* __restrict__ C)
{
    __shared__ float As[16 * 128];            // 8 KB A tile, row-major
    const int mbase    = blockIdx.x * 16;
    const int nthreads = 32 * (DN / 16);

    // cooperative coalesced load of the 16x128 fp32 A tile into LDS
    for (int i = threadIdx.x; i < 16 * 128 / 4; i += nthreads) {
        ((v4f*)As)[i] = ((const v4f*)(A + (size_t)mbase * 128))[i];
    }
    __syncthreads();

    const int lane = threadIdx.x & 31;
    const int wave = threadIdx.x >> 5;
    const int lo16 = lane & 15;               // row (A) / column (B,C)
    const int hi   = lane >> 4;               // selects K pair 0/1 vs 2/3, C row half
    const int col  = wave * 16 + lo16;        // global output column

    v8f c = {};
#pragma unroll
    for (int kk = 0; kk < 128; kk += 4) {
        // A fragment: lane holds A[m=lo16][kk + hi*2 + {0,1}]  (8B-aligned LDS read)
        v2f a = *(const v2f*)(&As[lo16 * 128 + kk + hi * 2]);
        // B fragment: lane holds W[kk + hi*2 + {0,1}][col]
        v2f b;
        b.x = W[(kk + hi * 2 + 0) * DN + col];
        b.y = W[(kk + hi * 2 + 1) * DN + col];
        // D = A(16x4) * B(4x16) + C  (fp32, exact path)
        c = __builtin_amdgcn_wmma_f32_16x16x4_f32(
            /*neg_a=*/false, a, /*neg_b=*/false, b,
            /*c_mod=*/(short)0, c, /*reuse_a=*/false, /*reuse_b=*/false);
    }

    // C/D layout: lanes 0-15 rows 0-7 in VGPR 0-7; lanes 16-31 rows 8-15
#pragma unroll
    for (int r = 0; r < 8; ++r) {
        C[(size_t)(mbase + hi * 8 + r) * DN + col] = c[r];
    }
}

// ---------------------------------------------------------------------------
// Edge scatter: out[dst[e]][:] += M[src[e]][:], D floats per edge.
// One thread per (edge, 4-float chunk): a wave covers one edge for D=128,
// giving a contiguous float4 gather + coalesced f32 atomics into L2.
// ---------------------------------------------------------------------------
template <int D>
__global__ __launch_bounds__(256) void scatter_add(
    const float* __restrict__ M, const int* __restrict__ ei,
    float* __restrict__ out)
{
    const long long total = (long long)N_EDGES * (D / 4);
    long long gid = (long long)blockIdx.x * blockDim.x + threadIdx.x;
    if (gid >= total) return;
    const int e  = (int)(gid / (D / 4));
    const int c4 = (int)(gid % (D / 4));
    const int s  = ei[e];                 // edge_index[0][e]
    const int d  = ei[N_EDGES + e];       // edge_index[1][e]
    const float* srcp = M + (size_t)s * D + c4 * 4;
    float*       dstp = out + (size_t)d * D + c4 * 4;
    v4f v = *(const v4f*)srcp;
#pragma unroll
    for (int j = 0; j < 4; ++j) {
        __hip_atomic_fetch_add(dstp + j, v[j],
                               __ATOMIC_RELAXED, __HIP_MEMORY_SCOPE_AGENT);
    }
}

// h = relu(gamma*(h + b1 - rm)*rsqrt(rv+eps) + beta), in place, [N, 128]
__global__ __launch_bounds__(256) void bias_bn_relu(
    float* __restrict__ h, const float* __restrict__ b1,
    const float* __restrict__ gamma, const float* __restrict__ beta,
    const float* __restrict__ rm, const float* __restrict__ rv)
{
    long long i = (long long)blockIdx.x * blockDim.x + threadIdx.x;
    if (i >= (long long)N_NODES * D_HID) return;
    const int c = (int)(i & (D_HID - 1));
    float v = h[i] + b1[c];
    v = gamma[c] * (v - rm[c]) * rsqrtf(rv[c] + BN_EPS) + beta[c];
    h[i] = v > 0.f ? v : 0.f;
}

__global__ __launch_bounds__(256) void zero_buf(float* __restrict__ p, long long n)
{
    long long i = (long long)blockIdx.x * blockDim.x + threadIdx.x;
    if (i < n) p[i] = 0.f;
}

__global__ __launch_bounds__(256) void init_bias(
    float* __restrict__ out, const float* __restrict__ b, long long n)
{
    long long i = (long long)blockIdx.x * blockDim.x + threadIdx.x;
    if (i < n) out[i] = b[(int)(i & (D_OUT - 1))];
}

extern "C" void kernel_launch(void* const* d_in, const int* in_sizes, int n_in,
                              void* d_out, int out_size, void* d_ws, size_t ws_size,
                              hipStream_t stream)
{
    const float* x      = (const float*)d_in[0];
    const float* W1     = (const float*)d_in[1];
    const float* b1     = (const float*)d_in[2];
    const float* gamma1 = (const float*)d_in[3];
    const float* beta1  = (const float*)d_in[4];
    const float* rm1    = (const float*)d_in[5];
    const float* rv1    = (const float*)d_in[6];
    const float* W2     = (const float*)d_in[7];
    const float* b2     = (const float*)d_in[8];
    const int*   ei     = (const int*)d_in[9];
    float* out = (float*)d_out;

    float* m   = (float*)d_ws;                         // [N, 128] (reused as [N, 64])
    float* agg = m + (size_t)N_NODES * D_HID;          // [N, 128]

    // 1) zero aggregation buffer (fresh every call)
    {
        long long n = (long long)N_NODES * D_HID;
        zero_buf<<<(int)((n + 255) / 256), 256, 0, stream>>>(agg, n);
    }
    // 2) m = x @ W1   (fp32 WMMA)
    gemm_f32_wmma<128><<<N_NODES / 16, 256, 0, stream>>>(x, W1, m);
    // 3) agg[dst] += m[src]  (128-wide)
    {
        long long t = (long long)N_EDGES * 32;
        scatter_add<128><<<(int)((t + 255) / 256), 256, 0, stream>>>(m, ei, agg);
    }
    // 4) agg = relu(BN(agg + b1)) in place
    {
        long long n = (long long)N_NODES * D_HID;
        bias_bn_relu<<<(int)((n + 255) / 256), 256, 0, stream>>>(agg, b1, gamma1, beta1, rm1, rv1);
    }
    // 5) m = agg @ W2  (fp32 WMMA, 64-wide output)
    gemm_f32_wmma<64><<<N_NODES / 16, 128, 0, stream>>>(agg, W2, m);
    // 6) out = b2 (broadcast)
    {
        long long n = (long long)N_NODES * D_OUT;
        init_bias<<<(int)((n + 255) / 256), 256, 0, stream>>>(out, b2, n);
    }
    // 7) out[dst] += m[src]  (64-wide)
    {
        long long t = (long long)N_EDGES * 16;
        scatter_add<64><<<(int)((t + 255) / 256), 256, 0, stream>>>(m, ei, out);
    }
}